// EntropyModelSoS_61589831024666
// MI455X (gfx1250) — compile-verified
//
#include <hip/hip_runtime.h>
#include <stdint.h>

// EntropyModelSoS quantize(symbols)+dequantize for MI455X (gfx1250).
// HBM-bound (12 B/elem, ~151 MB -> ~6.5us floor @ 23.3 TB/s). No matrix ops,
// so no WMMA; CDNA5 paths used: async global->LDS DMA (ASYNCcnt) for the
// level/mean tables, wave32 B128 vector loads/stores, conflict-free DS gathers.

namespace {

constexpr int kLvl = 64;   // quantization levels
constexpr int kCh  = 192;  // channels (means)

// Branchless lower-bound over a sorted 64-entry LDS array (m[63] == +inf):
// returns count of m[j] < x, i.e. jnp.searchsorted(mid, x) for 63 real mids.
__device__ __forceinline__ int lower_count(const float* __restrict__ m, float x) {
  int b = 0;
  b += (m[b + 31] < x) ? 32 : 0;
  b += (m[b + 15] < x) ? 16 : 0;
  b += (m[b +  7] < x) ?  8 : 0;
  b += (m[b +  3] < x) ?  4 : 0;
  b += (m[b +  1] < x) ?  2 : 0;
  b += (m[b     ] < x) ?  1 : 0;
  return b;
}

__global__ __launch_bounds__(256) void sos_quant_dequant(
    const float* __restrict__ x,
    const float* __restrict__ levels,
    const float* __restrict__ means,
    int*   __restrict__ sym_out,
    float* __restrict__ deq_out,
    int nv4)  // N/4 float4 elements
{
  __shared__ float s_lvl[kLvl];   // sorted levels
  __shared__ float s_mid[kLvl];   // 63 midpoints + inf pad
  __shared__ float s_mean[kCh];   // per-channel means

  const int t = (int)threadIdx.x;

  // ---- Stage tables with CDNA5 async global->LDS DMA --------------------
  // 256 lanes = 8 waves; waves 0-1 pull the 64 levels, waves 2-7 pull the
  // 192 means (saddr is wave-uniform in each branch). The flat address of a
  // __shared__ object has the LDS byte offset in its low 32 bits (aperture
  // hi bits live in [63:32]), which is exactly what VDST wants.
  if (t < kLvl) {
    unsigned ldsa = (unsigned)(uintptr_t)(&s_lvl[t]);
    unsigned off  = (unsigned)t * 4u;
    asm volatile("global_load_async_to_lds_b32 %0, %1, %2"
                 :: "v"(ldsa), "v"(off), "s"(levels) : "memory");
  } else {
    unsigned ldsa = (unsigned)(uintptr_t)(&s_mean[t - kLvl]);
    unsigned off  = (unsigned)(t - kLvl) * 4u;
    asm volatile("global_load_async_to_lds_b32 %0, %1, %2"
                 :: "v"(ldsa), "v"(off), "s"(means) : "memory");
  }
  asm volatile("s_wait_asynccnt 0" ::: "memory");  // our wave's DMA landed in LDS
  __syncthreads();                                  // visible to all 8 waves

  // Decision boundaries: mid[j] = (lv[j]+lv[j+1])/2, padded with +inf.
  if (t < kLvl - 1) {
    s_mid[t] = 0.5f * (s_lvl[t] + s_lvl[t + 1]);
  } else if (t == kLvl - 1) {
    s_mid[t] = __builtin_inff();
  }
  __syncthreads();

  // ---- Streaming phase: B128 in, 2x B128 out per vec4 -------------------
  // H*W = 4096 elems/channel => 1024 vec4 per channel; a vec4 never straddles
  // channels, so one mean lookup per vec4.
  const int stride = (int)(gridDim.x * blockDim.x);
  for (int i = (int)(blockIdx.x * blockDim.x) + t; i < nv4; i += stride) {
    const float4 xv  = reinterpret_cast<const float4*>(x)[i];
    const int   c    = (i >> 10) % kCh;
    const float mean = s_mean[c];

    int4 sv; float4 dv;
    sv.x = lower_count(s_mid, xv.x); dv.x = s_lvl[sv.x] + mean;
    sv.y = lower_count(s_mid, xv.y); dv.y = s_lvl[sv.y] + mean;
    sv.z = lower_count(s_mid, xv.z); dv.z = s_lvl[sv.z] + mean;
    sv.w = lower_count(s_mid, xv.w); dv.w = s_lvl[sv.w] + mean;

    reinterpret_cast<int4*>(sym_out)[i]   = sv;   // global_store_b128
    reinterpret_cast<float4*>(deq_out)[i] = dv;   // global_store_b128
  }
}

}  // namespace

extern "C" void kernel_launch(void* const* d_in, const int* in_sizes, int n_in,
                              void* d_out, int out_size, void* d_ws, size_t ws_size,
                              hipStream_t stream) {
  (void)n_in; (void)out_size; (void)d_ws; (void)ws_size;

  const float* x      = (const float*)d_in[0];  // [B,C,H,W] f32
  const float* levels = (const float*)d_in[1];  // [64] f32 (sorted)
  const float* means  = (const float*)d_in[2];  // [192] f32

  const int n   = in_sizes[0];   // 12,582,912 (multiple of 4)
  const int nv4 = n >> 2;

  int*   sym = (int*)d_out;          // output 0: symbols (int32), N elems
  float* deq = (float*)d_out + n;    // output 1: dequantized (f32), N elems

  // ~4 vec4 per thread via grid-stride; plenty of blocks to cover all WGPs.
  int blocks = (nv4 + 256 * 4 - 1) / (256 * 4);
  if (blocks < 1) blocks = 1;
  sos_quant_dequant<<<blocks, 256, 0, stream>>>(x, levels, means, sym, deq, nv4);
}